// GIN_87462714015856
// MI455X (gfx1250) — compile-verified
//
#include <hip/hip_runtime.h>
#include <hip/hip_bf16.h>

// ---------------------------------------------------------------------------
// GINEConv GNN for MI455X (gfx1250, wave32).
//   - Node/head GEMMs ([*,64] @ [64,64]) via V_WMMA_F32_16X16X4_F32 (exact fp32)
//   - B fragments staged in LDS with k-pairs interleaved per column so each
//     lane's {W[kb][col], W[kb+1][col]} is one aligned ds_load_b64 -> feeds the
//     WMMA source pair directly (no VGPR repacking movs).
//   - BN column stats fused into GEMM epilogue (atomics), BN+relu fused into
//     the next GEMM's A-operand load.
//   - Edge path: per-layer folded edge transform W' = w_edge @ lin_w[l],
//     one wave32 per edge, __shfl broadcast of 16 edge features,
//     global_atomic_add_f32 scatter into agg.
// ---------------------------------------------------------------------------

typedef float v2f __attribute__((ext_vector_type(2)));
typedef float v8f __attribute__((ext_vector_type(8)));

#define NN      100000
#define EE      1200000
#define DD      64
#define LL      3
#define GG      512
#define CC      10
#define BN_EPS  1e-5f

// ---------------------------------------------------------------------------
// zero fill
// ---------------------------------------------------------------------------
__global__ void k_zero(float* __restrict__ p, int n) {
    int i = blockIdx.x * blockDim.x + threadIdx.x;
    if (i < n) p[i] = 0.0f;
}

// ---------------------------------------------------------------------------
// Fold edge encoder + per-layer lin into W'[l] (16x64) and b'[l] (64).
//   W'[l] = w_edge(16x64) @ lin_w[l](64x64)
//   b'[l] = b_edge @ lin_w[l] + lin_b[l]
// ---------------------------------------------------------------------------
__global__ void k_prep_edge(const float* __restrict__ w_edge,
                            const float* __restrict__ b_edge,
                            const float* __restrict__ lin_w,
                            const float* __restrict__ lin_b,
                            float* __restrict__ Wp, float* __restrict__ bp) {
    int idx = blockIdx.x * blockDim.x + threadIdx.x;
    if (idx < LL * 16 * 64) {
        int l = idx >> 10; int rem = idx & 1023;
        int r = rem >> 6;  int c = rem & 63;
        const float* lw = lin_w + l * 4096;
        float acc = 0.0f;
        #pragma unroll 8
        for (int i = 0; i < 64; ++i) acc = fmaf(w_edge[r * 64 + i], lw[i * 64 + c], acc);
        Wp[idx] = acc;
    } else {
        int j = idx - LL * 16 * 64;
        if (j < LL * 64) {
            int l = j >> 6; int c = j & 63;
            const float* lw = lin_w + l * 4096;
            float acc = lin_b[l * 64 + c];
            #pragma unroll 8
            for (int i = 0; i < 64; ++i) acc = fmaf(b_edge[i], lw[i * 64 + c], acc);
            bp[j] = acc;
        }
    }
}

// ---------------------------------------------------------------------------
// Edge message + scatter-add.  One wave32 per edge, 2 output dims per lane.
//   m = relu(h[src] + edge_attr[e](1x16) @ Wp(16x64) + bp)
//   agg[dst] += m   (global f32 atomics)
// ---------------------------------------------------------------------------
__global__ void k_edge(const float* __restrict__ h,
                       const float* __restrict__ edge_attr,
                       const int* __restrict__ src, const int* __restrict__ dst,
                       const float* __restrict__ Wp, const float* __restrict__ bp,
                       float* __restrict__ agg, int E) {
    const int lane   = threadIdx.x & 31;
    const int wave   = (blockIdx.x * blockDim.x + threadIdx.x) >> 5;
    const int nwaves = (gridDim.x * blockDim.x) >> 5;
    const int d0 = lane * 2;

    float2 wp[16];
    #pragma unroll
    for (int k = 0; k < 16; ++k)
        wp[k] = make_float2(Wp[k * 64 + d0], Wp[k * 64 + d0 + 1]);
    const float2 bb = make_float2(bp[d0], bp[d0 + 1]);

    for (int e = wave; e < E; e += nwaves) {
        const int s = src[e];
        const int d = dst[e];
        const float av = edge_attr[e * 16 + (lane & 15)];
        const float2 hv = *(const float2*)(h + s * 64 + d0);
        float mx = bb.x, my = bb.y;
        #pragma unroll
        for (int k = 0; k < 16; ++k) {
            const float a = __shfl(av, k, 32);   // wave32 broadcast
            mx = fmaf(a, wp[k].x, mx);
            my = fmaf(a, wp[k].y, my);
        }
        mx = fmaxf(mx + hv.x, 0.0f);
        my = fmaxf(my + hv.y, 0.0f);
        atomicAdd(agg + d * 64 + d0,     mx);
        atomicAdd(agg + d * 64 + d0 + 1, my);
    }
}

// ---------------------------------------------------------------------------
// WMMA GEMM: OUT[N,64] = post( pre(A)[N,64] @ W[64,64] + bias )
//   PRE  0: A0            1: A0 + A1        2: relu(psc[k]*A0 + psh[k])
//   POST 0: none          1: relu
//   STATS: accumulate per-column sum / sumsq of OUT (for training-mode BN)
// One wave32 computes a 16-row x 64-col tile with 64 x v_wmma_f32_16x16x4_f32.
// ---------------------------------------------------------------------------
template <int PRE>
__device__ __forceinline__ v2f load_a(const float* __restrict__ A0,
                                      const float* __restrict__ A1,
                                      const float* __restrict__ psc,
                                      const float* __restrict__ psh,
                                      int row, int kb) {
    const float2 v = *(const float2*)(A0 + row * 64 + kb);
    v2f a;
    if constexpr (PRE == 0) {
        a.x = v.x; a.y = v.y;
    } else if constexpr (PRE == 1) {
        const float2 u = *(const float2*)(A1 + row * 64 + kb);
        a.x = v.x + u.x; a.y = v.y + u.y;
    } else {
        const float2 sc = *(const float2*)(psc + kb);
        const float2 sh = *(const float2*)(psh + kb);
        a.x = fmaxf(fmaf(sc.x, v.x, sh.x), 0.0f);
        a.y = fmaxf(fmaf(sc.y, v.y, sh.y), 0.0f);
    }
    return a;
}

template <int PRE, int POST, bool STATS>
__global__ __launch_bounds__(128)
void k_gemm64(const float* __restrict__ A0, const float* __restrict__ A1,
              const float* __restrict__ psc, const float* __restrict__ psh,
              const float* __restrict__ W, const float* __restrict__ bias,
              float* __restrict__ OUT,
              float* __restrict__ colsum, float* __restrict__ colsumsq,
              int Nrows) {
    constexpr int WAVES = 4;                  // 128 threads = 4 wave32
    // LDS layout: k-pairs interleaved per column:
    //   Wl[(k>>1)*128 + col*2 + (k&1)]  ->  {W[2j][col], W[2j+1][col]} adjacent
    __shared__ float Wl[64 * 64];
    __shared__ float Bl[64];
    const int tid = threadIdx.x;
    for (int i = tid; i < 4096; i += 128) {
        const int k = i >> 6, col = i & 63;
        Wl[(k >> 1) * 128 + col * 2 + (k & 1)] = W[i];
    }
    if (tid < 64) Bl[tid] = bias[tid];
    __syncthreads();

    const int lane = tid & 31;
    const int wv   = tid >> 5;
    const int row0 = (blockIdx.x * WAVES + wv) * 16;
    if (row0 >= Nrows) return;
    const int half = lane >> 4;               // K split across half-waves
    const int m    = lane & 15;
    const int row  = row0 + m;                // A-fragment row for this lane

    v8f acc[4] = {};                          // 4 col-tiles of 16
    #pragma unroll
    for (int kc = 0; kc < 16; ++kc) {         // K = 64 in chunks of 4
        const int kb = kc * 4 + half * 2;     // even
        const v2f a = load_a<PRE>(A0, A1, psc, psh, row, kb);
        const float* wrow = &Wl[(kb >> 1) * 128];   // 128 floats: 64 cols x 2
        #pragma unroll
        for (int ct = 0; ct < 4; ++ct) {
            const v2f b = *(const v2f*)(wrow + (ct * 16 + m) * 2);  // ds_load_b64
            acc[ct] = __builtin_amdgcn_wmma_f32_16x16x4_f32(
                false, a, false, b, (short)0, acc[ct], false, false);
        }
    }

    // Epilogue: bias (+relu), store, optional per-column BN stats.
    // C layout: VGPR i -> row (row0 + i + half*8), col = ct*16 + (lane&15).
    #pragma unroll
    for (int ct = 0; ct < 4; ++ct) {
        const int col = ct * 16 + m;
        const float bs = Bl[col];
        float ssum = 0.0f, ssq = 0.0f;
        #pragma unroll
        for (int i = 0; i < 8; ++i) {
            const int r = row0 + half * 8 + i;
            float v = acc[ct][i] + bs;
            if constexpr (POST == 1) v = fmaxf(v, 0.0f);
            OUT[r * 64 + col] = v;
            if constexpr (STATS) { ssum += v; ssq = fmaf(v, v, ssq); }
        }
        if constexpr (STATS) {
            atomicAdd(&colsum[col], ssum);
            atomicAdd(&colsumsq[col], ssq);
        }
    }
}

// ---------------------------------------------------------------------------
// BN stats -> per-column scale/shift.  (biased variance, training mode)
// ---------------------------------------------------------------------------
__global__ void k_bnstats(const float* __restrict__ colsum,
                          const float* __restrict__ colsumsq,
                          const float* __restrict__ g, const float* __restrict__ b,
                          float* __restrict__ scale, float* __restrict__ shift,
                          float invN) {
    const int i = threadIdx.x;                // 64 threads
    const float mean = colsum[i] * invN;
    const float var  = colsumsq[i] * invN - mean * mean;
    const float sc   = g[i] * rsqrtf(var + BN_EPS);
    scale[i] = sc;
    shift[i] = fmaf(-mean, sc, b[i]);
}

// ---------------------------------------------------------------------------
// global_add_pool: pooled[batch[n]] += h[n]
// ---------------------------------------------------------------------------
__global__ void k_pool(const float* __restrict__ h, const int* __restrict__ batch,
                       float* __restrict__ pooled, int N) {
    const int idx = blockIdx.x * blockDim.x + threadIdx.x;
    if (idx >= N * 64) return;
    const int n = idx >> 6;
    const int d = idx & 63;
    atomicAdd(pooled + batch[n] * 64 + d, h[idx]);
}

// ---------------------------------------------------------------------------
// Final head: out[g,c] = bn2(t2)[g,:] @ fo3[:,c] + b3[c]
// ---------------------------------------------------------------------------
__global__ void k_head(const float* __restrict__ t2,
                       const float* __restrict__ scale, const float* __restrict__ shift,
                       const float* __restrict__ fo3_w, const float* __restrict__ fo3_b,
                       float* __restrict__ out) {
    const int idx = blockIdx.x * blockDim.x + threadIdx.x;
    if (idx >= GG * CC) return;
    const int g = idx / CC;
    const int c = idx - g * CC;
    float v = fo3_b[c];
    #pragma unroll 8
    for (int k = 0; k < 64; ++k) {
        const float t = fmaf(scale[k], t2[g * 64 + k], shift[k]);
        v = fmaf(t, fo3_w[k * CC + c], v);
    }
    out[idx] = v;
}

// ---------------------------------------------------------------------------
extern "C" void kernel_launch(void* const* d_in, const int* in_sizes, int n_in,
                              void* d_out, int out_size, void* d_ws, size_t ws_size,
                              hipStream_t stream) {
    const float* x         = (const float*)d_in[0];
    const int*   edge_idx  = (const int*)  d_in[1];
    const int*   batch     = (const int*)  d_in[2];
    const float* edge_attr = (const float*)d_in[3];
    const float* w_node    = (const float*)d_in[4];
    const float* b_node    = (const float*)d_in[5];
    const float* w_edge    = (const float*)d_in[6];
    const float* b_edge    = (const float*)d_in[7];
    const float* lin_w     = (const float*)d_in[8];
    const float* lin_b     = (const float*)d_in[9];
    const float* mlp1_w    = (const float*)d_in[10];
    const float* mlp1_b    = (const float*)d_in[11];
    const float* bn1_g     = (const float*)d_in[12];
    const float* bn1_b     = (const float*)d_in[13];
    const float* mlp2_w    = (const float*)d_in[14];
    const float* mlp2_b    = (const float*)d_in[15];
    const float* fo1_w     = (const float*)d_in[16];
    const float* fo1_b     = (const float*)d_in[17];
    const float* fo_bn1_g  = (const float*)d_in[18];
    const float* fo_bn1_b  = (const float*)d_in[19];
    const float* fo2_w     = (const float*)d_in[20];
    const float* fo2_b     = (const float*)d_in[21];
    const float* fo_bn2_g  = (const float*)d_in[22];
    const float* fo_bn2_b  = (const float*)d_in[23];
    const float* fo3_w     = (const float*)d_in[24];
    const float* fo3_b     = (const float*)d_in[25];
    float* out = (float*)d_out;

    const int* src = edge_idx;           // edge_index[0]
    const int* dst = edge_idx + EE;      // edge_index[1]

    // ---- workspace carve-up (floats) ----
    float* ws = (float*)d_ws;
    const int N64 = NN * 64;             // 6,400,000
    float* h        = ws;                // [N,64]
    float* agg      = ws + (size_t)N64;  // [N,64]
    float* t        = ws + (size_t)2 * N64;          // [N,64]
    float* Wp       = ws + (size_t)3 * N64;          // 3*16*64
    float* bp       = Wp + 3 * 16 * 64;              // 3*64
    float* colsum   = bp + 3 * 64;                   // 64
    float* colsumsq = colsum + 64;                   // 64 (contiguous w/ colsum)
    float* scale    = colsumsq + 64;                 // 64
    float* shift    = scale + 64;                    // 64
    float* pooled   = shift + 64;                    // 512*64
    float* t1       = pooled + GG * 64;              // 512*64
    float* t2       = t1 + GG * 64;                  // 512*64

    const int gemm_grid_n = (NN / 16 + 3) / 4;       // 4 wave-tiles / block
    const int gemm_grid_g = (GG / 16 + 3) / 4;
    const float invN = 1.0f / (float)NN;
    const float invG = 1.0f / (float)GG;

    // fold edge encoder + per-layer lin
    k_prep_edge<<<(LL * 16 * 64 + LL * 64 + 255) / 256, 256, 0, stream>>>(
        w_edge, b_edge, lin_w, lin_b, Wp, bp);

    // node encoder: h = x @ w_node + b_node
    k_gemm64<0, 0, false><<<gemm_grid_n, 128, 0, stream>>>(
        x, nullptr, nullptr, nullptr, w_node, b_node, h, nullptr, nullptr, NN);

    for (int l = 0; l < LL; ++l) {
        k_zero<<<(N64 + 255) / 256, 256, 0, stream>>>(agg, N64);
        k_zero<<<1, 128, 0, stream>>>(colsum, 128);  // colsum+colsumsq

        k_edge<<<1024, 256, 0, stream>>>(h, edge_attr, src, dst,
                                         Wp + l * 1024, bp + l * 64, agg, EE);

        // t = (agg + h) @ mlp1 + b1, with BN stats
        k_gemm64<1, 0, true><<<gemm_grid_n, 128, 0, stream>>>(
            agg, h, nullptr, nullptr, mlp1_w + l * 4096, mlp1_b + l * 64,
            t, colsum, colsumsq, NN);

        k_bnstats<<<1, 64, 0, stream>>>(colsum, colsumsq,
                                        bn1_g + l * 64, bn1_b + l * 64,
                                        scale, shift, invN);

        // h = relu( relu(bn(t)) @ mlp2 + b2 )
        k_gemm64<2, 1, false><<<gemm_grid_n, 128, 0, stream>>>(
            t, nullptr, scale, shift, mlp2_w + l * 4096, mlp2_b + l * 64,
            h, nullptr, nullptr, NN);
    }

    // global_add_pool
    k_zero<<<(GG * 64 + 255) / 256, 256, 0, stream>>>(pooled, GG * 64);
    k_pool<<<(N64 + 255) / 256, 256, 0, stream>>>(h, batch, pooled, NN);

    // head: t1 = pooled @ fo1 + b, stats
    k_zero<<<1, 128, 0, stream>>>(colsum, 128);
    k_gemm64<0, 0, true><<<gemm_grid_g, 128, 0, stream>>>(
        pooled, nullptr, nullptr, nullptr, fo1_w, fo1_b, t1, colsum, colsumsq, GG);
    k_bnstats<<<1, 64, 0, stream>>>(colsum, colsumsq, fo_bn1_g, fo_bn1_b,
                                    scale, shift, invG);

    // t2 = relu(bn(t1)) @ fo2 + b, stats
    k_zero<<<1, 128, 0, stream>>>(colsum, 128);
    k_gemm64<2, 0, true><<<gemm_grid_g, 128, 0, stream>>>(
        t1, nullptr, scale, shift, fo2_w, fo2_b, t2, colsum, colsumsq, GG);
    k_bnstats<<<1, 64, 0, stream>>>(colsum, colsumsq, fo_bn2_g, fo_bn2_b,
                                    scale, shift, invG);

    // logits = bn(t2) @ fo3 + b3
    k_head<<<(GG * CC + 255) / 256, 256, 0, stream>>>(
        t2, scale, shift, fo3_w, fo3_b, out);
}